// BendingEnergy_7310034338162
// MI455X (gfx1250) — compile-verified
//
#include <hip/hip_runtime.h>
#include <hip/hip_bf16.h>
#include <stdint.h>

// ---------------- problem constants (reference: (4,160,160,160,3) f32) ----
#define NPT   160
#define NC    3
#define SH    (NPT * NC)            // h-stride in elements = 480
#define SD    (NPT * NPT * NC)      // d-stride = 76800
#define SB    (NPT * NPT * NPT * NC)// batch stride = 12,288,000
#define TOUT  12                    // output tile edge (156 = 12 * 13)
#define NT    13                    // tiles per axis
#define DDIM  16                    // input tile depth  (TOUT + 4 halo)
#define HDIM  16                    // input tile height
#define WDIM  16                    // input tile width
#define XDIM  (WDIM * NC)           // flattened (w,c) row = 48 elements
#define LDS_ELEMS (DDIM * HDIM * XDIM)  // 12288 floats = 48 KB
#define BLK   256
#define PTS   (TOUT * TOUT * TOUT)  // 1728 output points per tile
#define TILES_PER_BATCH (NT * NT * NT)  // 2197
#define INV_COUNT (1.0f / 11389248.0f)  // 1 / (156^3 * 3)

typedef float v2f  __attribute__((ext_vector_type(2)));
typedef float v4f  __attribute__((ext_vector_type(4)));
typedef float v8f  __attribute__((ext_vector_type(8)));
typedef unsigned int u32x4 __attribute__((ext_vector_type(4)));
typedef int   i32x4 __attribute__((ext_vector_type(4)));
typedef int   i32x8 __attribute__((ext_vector_type(8)));

// Hard proof that the DEVICE binary uses the TDM path: the device pass must
// have the builtin or the build fails. (The host pass never declares
// __builtin_amdgcn_*, so it must be excluded from this check.)
#if defined(__HIP_DEVICE_COMPILE__) && !__has_builtin(__builtin_amdgcn_tensor_load_to_lds)
#error "TDM builtin __builtin_amdgcn_tensor_load_to_lds not available in device pass"
#endif

__global__ __launch_bounds__(BLK)
void be_pass1(const float* __restrict__ f, float* __restrict__ partial) {
  __shared__ float tile[LDS_ELEMS];
  __shared__ float wsum[BLK / 32];

  const int bx  = blockIdx.x;           // w tile
  const int by  = blockIdx.y;           // h tile
  const int bz  = blockIdx.z % NT;      // d tile
  const int b   = blockIdx.z / NT;      // batch
  const int tid = threadIdx.x;

  // Element offset of the input tile origin. Output origin is 2 + bt*12, so
  // the input origin (with -2 halo) is bt*12; halo never leaves [0,160).
  const long long goff = (long long)b * SB + (long long)(bz * TOUT) * SD
                       + (long long)(by * TOUT) * SH + (long long)(bx * TOUT) * NC;

#if defined(__HIP_DEVICE_COMPILE__)
  // ---- TDM: one DMA moves the 16x16x(16w x 3c)-element tile into LDS ------
  // Issued by a single wave (TDM ignores EXEC; the branch keeps the other 7
  // waves from duplicating the DMA), tracked on TENSORcnt.
  if (tid < 32) {
    const uint64_t ga      = (uint64_t)(uintptr_t)(f + goff);
    const uint32_t lds_off = (uint32_t)(uintptr_t)&tile[0];  // low 32b of generic LDS addr

    u32x4 g0;
    g0[0] = 1u;                                        // count=1, user descriptor
    g0[1] = lds_off;                                   // lds_addr (bytes)
    g0[2] = (uint32_t)ga;                              // global_addr[31:0]
    g0[3] = (uint32_t)((ga >> 32) & 0x01FFFFFFull)     // global_addr[56:32]
          | (2u << 30);                                // type = 2 ("image")

    i32x8 g1;
    g1[0] = (2 << 16);                                 // data_size = 4 bytes
    g1[1] = (int)((uint32_t)(SH & 0xFFFF) << 16);      // tensor_dim0[15:0] = 480
    g1[2] = (int)(((uint32_t)SH >> 16)                 // tensor_dim0[31:16] = 0
          | ((uint32_t)(NPT & 0xFFFF) << 16));         // tensor_dim1[15:0] = 160
    g1[3] = (int)(((uint32_t)NPT >> 16)                // tensor_dim1[31:16] = 0
          | ((uint32_t)XDIM << 16));                   // tile_dim0 = 48 elements (192 B rows)
    g1[4] = (int)((uint32_t)HDIM | ((uint32_t)DDIM << 16)); // tile_dim1=16, tile_dim2=16
    g1[5] = SH;                                        // tensor_dim0_stride[31:0] = 480
    g1[6] = (int)(((uint32_t)(SD & 0xFFFF)) << 16);    // dim0_stride[47:32]=0 | dim1_stride[15:0]
    g1[7] = (int)((uint32_t)SD >> 16);                 // tensor_dim1_stride[47:16]

    i32x4 g2;
    g2[0] = NPT;                                       // tensor_dim2 = 160
    g2[1] = 0; g2[2] = 0; g2[3] = 0;                   // dim3 / stride2 / tile_dim3 unused
    i32x4 g3; g3[0] = 0; g3[1] = 0; g3[2] = 0; g3[3] = 0;

#if __clang_major__ >= 23
    i32x8 gpad; gpad[0]=0; gpad[1]=0; gpad[2]=0; gpad[3]=0;
    gpad[4]=0; gpad[5]=0; gpad[6]=0; gpad[7]=0;
    __builtin_amdgcn_tensor_load_to_lds(g0, g1, g2, g3, gpad, 0);
#else
    __builtin_amdgcn_tensor_load_to_lds(g0, g1, g2, g3, 0);
#endif
#if __has_builtin(__builtin_amdgcn_s_wait_tensorcnt)
    __builtin_amdgcn_s_wait_tensorcnt(0);
#else
    asm volatile("s_wait_tensorcnt 0x0" ::: "memory");
#endif
  }
#else
  // Host-pass parse body only (never codegen'd for the GPU): plain staging.
  for (int i = tid; i < LDS_ELEMS; i += BLK) {
    const int xx = i % XDIM;
    const int r  = i / XDIM;
    const int hh = r % HDIM;
    const int dd = r / HDIM;
    tile[i] = f[goff + (long long)dd * SD + hh * SH + xx];
  }
#endif
  __syncthreads();

  // ---- 19-point second-derivative stencil from LDS ------------------------
  const int sx = NC;            // +1 in w
  const int sy = XDIM;          // +1 in h
  const int sz = HDIM * XDIM;   // +1 in d

  float acc = 0.f;
  for (int p = tid; p < PTS; p += BLK) {
    const int wx = p % TOUT;
    const int hy = (p / TOUT) % TOUT;
    const int dz = p / (TOUT * TOUT);
    const int base = ((dz + 2) * HDIM + (hy + 2)) * XDIM + (wx + 2) * NC;
#pragma unroll
    for (int c = 0; c < NC; ++c) {
      const int i0 = base + c;
      const float fc  = tile[i0];
      const float fxx = tile[i0 + 2*sz] - 2.f*fc + tile[i0 - 2*sz];
      const float fyy = tile[i0 + 2*sy] - 2.f*fc + tile[i0 - 2*sy];
      const float fzz = tile[i0 + 2*sx] - 2.f*fc + tile[i0 - 2*sx];
      const float fxy = tile[i0 + sz + sy] - tile[i0 - sz + sy]
                      - tile[i0 + sz - sy] + tile[i0 - sz - sy];
      const float fxz = tile[i0 + sz + sx] - tile[i0 - sz + sx]
                      - tile[i0 + sz - sx] + tile[i0 - sz - sx];
      const float fyz = tile[i0 + sy + sx] - tile[i0 - sy + sx]
                      - tile[i0 + sy - sx] + tile[i0 - sy - sx];
      acc += fxx*fxx + fyy*fyy + fzz*fzz + 2.f*(fxy*fxy + fxz*fxz + fyz*fyz);
    }
  }
  acc *= 0.0625f;   // common (0.25)^2 factor of every squared second derivative

  // ---- wave-level reduction via WMMA (A = partials, B = ones, C = 0) ------
  // D[m,n] = s[m] + s[m+16]; lane L holds rows r+(L<16?0:8) of D in 8 VGPRs,
  // so summing them yields a half-wave sum; one xor-16 shuffle finishes it.
  float t;
#if defined(__HIP_DEVICE_COMPILE__) && __has_builtin(__builtin_amdgcn_wmma_f32_16x16x4_f32)
  {
    v2f a;  a[0] = acc; a[1] = 0.f;
    v2f bo; bo[0] = 1.f; bo[1] = 1.f;     // all-ones B: layout-invariant
    v8f cz; cz[0]=0.f; cz[1]=0.f; cz[2]=0.f; cz[3]=0.f;
    cz[4]=0.f; cz[5]=0.f; cz[6]=0.f; cz[7]=0.f;
    v8f d = __builtin_amdgcn_wmma_f32_16x16x4_f32(false, a, false, bo,
                                                  (short)0, cz, false, false);
    t = ((d[0] + d[1]) + (d[2] + d[3])) + ((d[4] + d[5]) + (d[6] + d[7]));
    t += __shfl_xor(t, 16, 32);
  }
#else
  t = acc;
  for (int m = 1; m < 32; m <<= 1) t += __shfl_xor(t, m, 32);
#endif

  const int lane = tid & 31, wv = tid >> 5;
  if (lane == 0) wsum[wv] = t;
  __syncthreads();
  if (tid == 0) {
    float tot = 0.f;
#pragma unroll
    for (int i = 0; i < BLK / 32; ++i) tot += wsum[i];   // fixed order: deterministic
    const int tileId = bx + NT * by + NT * NT * bz;
    partial[b * TILES_PER_BATCH + tileId] = tot;
  }
}

__global__ __launch_bounds__(256)
void be_pass2(const float* __restrict__ partial, float* __restrict__ out) {
  __shared__ float red[256];
  const int b = blockIdx.x, tid = threadIdx.x;
  const float* pb = partial + (long long)b * TILES_PER_BATCH;

  // 2197 partials = 549 float4 + 1 tail float; vectorized, fixed order.
  const int NV = TILES_PER_BATCH / 4;       // 549
  const v4f* pv = (const v4f*)pb;
  float s = 0.f;
  for (int i = tid; i < NV; i += 256) {
    v4f v = pv[i];
    s += ((v[0] + v[1]) + (v[2] + v[3]));
  }
  if (tid == 0) s += pb[TILES_PER_BATCH - 1];

  red[tid] = s;
  __syncthreads();
  for (int off = 128; off > 0; off >>= 1) {
    if (tid < off) red[tid] += red[tid + off];
    __syncthreads();
  }
  if (tid == 0) out[b] = red[0] * INV_COUNT;
}

extern "C" void kernel_launch(void* const* d_in, const int* in_sizes, int n_in,
                              void* d_out, int out_size, void* d_ws, size_t ws_size,
                              hipStream_t stream) {
  (void)in_sizes; (void)n_in; (void)ws_size;
  const float* f   = (const float*)d_in[0];
  float*       out = (float*)d_out;
  float*       prt = (float*)d_ws;            // 4 * 2197 floats, fully rewritten each call

  const int B = out_size;                     // 4 batches
  dim3 grid1(NT, NT, NT * B);
  be_pass1<<<grid1, BLK, 0, stream>>>(f, prt);
  be_pass2<<<B, 256, 0, stream>>>(prt, out);
}